// Clar_Loss_81269371174952
// MI455X (gfx1250) — compile-verified
//
#include <hip/hip_runtime.h>

#define TILE 64
#define HALO 2
#define IN_W (TILE + 2*HALO)      // 68 (input tile incl. halo 2)
#define S_W  (TILE + 2)           // 66 (s^2 region incl. halo 1)
#define S_S  68                   // padded LDS stride for s^2
#define IMG  1024
#define NIMG 48                   // 16 batch * 3 channels

typedef __attribute__((ext_vector_type(2))) float v2f;
typedef __attribute__((ext_vector_type(8))) float v8f;

#if __has_builtin(__builtin_amdgcn_global_load_async_to_lds_b32)
#define ASYNC_LDS 1
#else
#define ASYNC_LDS 0
#endif

__device__ __forceinline__ void wait_async_zero() {
#if ASYNC_LDS
#if __has_builtin(__builtin_amdgcn_s_wait_asynccnt)
    __builtin_amdgcn_s_wait_asynccnt(0);
#else
    asm volatile("s_wait_asynccnt 0" ::: "memory");
#endif
#endif
}

// Stage a 68x68 edge-clamped tile into LDS. Async path: GLOBAL_LOAD_ASYNC_TO_LDS_B32
// (memory -> LDS direct, ASYNCcnt-tracked, no VGPR staging). Caller must
// wait_async_zero() + __syncthreads() before consuming.
__device__ __forceinline__ void stage_tile(const float* __restrict__ src,
                                           float* __restrict__ sIn,
                                           int tx0, int ty0, int tid)
{
    for (int idx = tid; idx < IN_W * IN_W; idx += 256) {
        int ly = idx / IN_W, lx = idx - ly * IN_W;
        int gy = ty0 + ly - HALO; gy = gy < 0 ? 0 : (gy > IMG - 1 ? IMG - 1 : gy);
        int gx = tx0 + lx - HALO; gx = gx < 0 ? 0 : (gx > IMG - 1 ? IMG - 1 : gx);
        const float* gp = src + ((size_t)gy * IMG + gx);
#if ASYNC_LDS
        __builtin_amdgcn_global_load_async_to_lds_b32(
            (__attribute__((address_space(1))) int*)(uintptr_t)gp,   // global AS1 (flat==global numerically)
            (__attribute__((address_space(3))) int*)(sIn + idx),     // LDS AS3
            /*imm offset=*/0, /*cpol=*/0);
#else
        sIn[idx] = *gp;
#endif
    }
}

// SML -> s^2 over the 66x66 ring-extended region (zero outside image = conv zero-pad).
// (x+1)/2 pre-scale folded in here as s *= 0.5 (the +1 cancels in all differences).
__device__ __forceinline__ void sml_sq(const float* __restrict__ sIn,
                                       float* __restrict__ sS2,
                                       int tx0, int ty0, int tid)
{
    for (int p = tid; p < S_W * S_W; p += 256) {
        int sy = p / S_W, sx = p - sy * S_W;
        int gy = ty0 + sy - 1, gx = tx0 + sx - 1;
        float v = 0.f;
        if ((unsigned)gy < (unsigned)IMG && (unsigned)gx < (unsigned)IMG) {
            const float* q = &sIn[sy * IN_W + sx];
            float c = q[IN_W + 1];
            float ortho = fabsf(c - q[1])    + fabsf(c - q[2*IN_W + 1])
                        + fabsf(c - q[IN_W]) + fabsf(c - q[IN_W + 2]);
            float diag  = fabsf(c - q[0])    + fabsf(c - q[2*IN_W + 2])
                        + fabsf(c - q[2])    + fabsf(c - q[2*IN_W]);
            float s = 0.5f * fmaf(0.707f, diag, ortho);
            v = s * s;
        }
        sS2[sy * S_S + sx] = v;
    }
}

// ---------------------------------------------------------------------------
// Fused tile kernel with async-LDS software pipeline:
//   stage A -> sml^2 A -> [issue async stage B | conv A] -> sml^2 B -> conv B
// ---------------------------------------------------------------------------
__global__ __launch_bounds__(256) void nsml_loss_tile(
    const float* __restrict__ A, const float* __restrict__ B,
    float* __restrict__ partials)
{
    __shared__ float sIn[IN_W * IN_W];   // 18.5 KB (single buffer: dead between phases)
    __shared__ float sS2[S_W * S_S];     // 18.0 KB
    __shared__ float sRed[8];

    const int tid = threadIdx.x;
    const int tx0 = blockIdx.x * TILE;
    const int ty0 = blockIdx.y * TILE;
    const size_t ibase = (size_t)blockIdx.z * IMG * IMG;

    const int ox  = tid & (TILE - 1);     // output column within tile
    const int oy0 = (tid >> 6) * 16;      // 16-row strip per thread

    float accA[16];
    float lsum = 0.f;

    // ---- image A: stage + sml^2 ----
    stage_tile(A + ibase, sIn, tx0, ty0, tid);
    wait_async_zero();
    __syncthreads();

    sml_sq(sIn, sS2, tx0, ty0, tid);
    __syncthreads();                      // all reads of sIn done -> sIn reusable

    // ---- issue image B staging; overlap with conv(A) which touches only sS2 ----
    stage_tile(B + ibase, sIn, tx0, ty0, tid);

    // ---- conv A: separable [1,2,1]x[1,2,1]/16, rolling horizontal sums ----
    {
        const float* col = &sS2[oy0 * S_S + ox];
        float a0 = col[0], a1 = col[1], a2 = col[2];
        float hm2 = fmaf(2.f, a1, a0 + a2);
        const float* r1 = col + S_S;
        float b0 = r1[0], b1 = r1[1], b2 = r1[2];
        float hm1 = fmaf(2.f, b1, b0 + b2);
        #pragma unroll
        for (int r = 0; r < 16; ++r) {
            const float* rr = col + (r + 2) * S_S;
            float c0 = rr[0], c1 = rr[1], c2 = rr[2];
            float hc = fmaf(2.f, c1, c0 + c2);
            accA[r] = fmaf(2.f, hm1, hm2 + hc) * 0.0625f;
            hm2 = hm1; hm1 = hc;
        }
    }

    wait_async_zero();
    __syncthreads();                      // B tile in LDS; conv(A) reads of sS2 done

    // ---- image B: sml^2 ----
    sml_sq(sIn, sS2, tx0, ty0, tid);
    __syncthreads();

    // ---- conv B + squared-diff accumulate ----
    {
        const float* col = &sS2[oy0 * S_S + ox];
        float a0 = col[0], a1 = col[1], a2 = col[2];
        float hm2 = fmaf(2.f, a1, a0 + a2);
        const float* r1 = col + S_S;
        float b0 = r1[0], b1 = r1[1], b2 = r1[2];
        float hm1 = fmaf(2.f, b1, b0 + b2);
        #pragma unroll
        for (int r = 0; r < 16; ++r) {
            const float* rr = col + (r + 2) * S_S;
            float c0 = rr[0], c1 = rr[1], c2 = rr[2];
            float hc = fmaf(2.f, c1, c0 + c2);
            float o  = fmaf(2.f, hm1, hm2 + hc) * 0.0625f;
            float d  = accA[r] - o;
            lsum = fmaf(d, d, lsum);
            hm2 = hm1; hm1 = hc;
        }
    }

    // ---- deterministic block reduction ----
    float v = lsum;
    #pragma unroll
    for (int off = 16; off; off >>= 1) v += __shfl_down(v, off, 32);
    if ((tid & 31) == 0) sRed[tid >> 5] = v;
    __syncthreads();
    if (tid == 0) {
        float s = 0.f;
        #pragma unroll
        for (int i = 0; i < 8; ++i) s += sRed[i];
        int wg = (blockIdx.z * gridDim.y + blockIdx.y) * gridDim.x + blockIdx.x;
        partials[wg] = s;
    }
}

// ---------------------------------------------------------------------------
// Final reduction via V_WMMA_F32_16X16X4_F32 (verified lowering in round 1).
// A = ones(16x4), B = 4x16 data chunk, C accumulates: every row of D equals the
// column sums, so sum of c[0] across 32 lanes = 2 * grand total.
// ---------------------------------------------------------------------------
__global__ __launch_bounds__(32) void nsml_reduce_wmma(
    const float* __restrict__ p, int n, float* __restrict__ out, float scale)
{
    int lane = threadIdx.x;           // one wave32; EXEC all ones around the WMMAs
    v8f c = {};
    v2f a; a[0] = 1.f; a[1] = 1.f;    // ones A-matrix (layout-invariant)

    for (int base = lane * 2; base < n; base += 64) {   // n is a multiple of 64
        v2f b; b[0] = p[base]; b[1] = p[base + 1];
        c = __builtin_amdgcn_wmma_f32_16x16x4_f32(
                /*neg_a=*/false, a, /*neg_b=*/false, b,
                /*c_mod=*/(short)0, c, /*reuse_a=*/false, /*reuse_b=*/false);
    }

    float v = c[0];                    // VGPR0 of C/D: rows M=0 and M=8
    #pragma unroll
    for (int off = 16; off; off >>= 1) v += __shfl_down(v, off, 32);
    if (lane == 0) out[0] = v * 0.5f * scale;
}

// ---------------------------------------------------------------------------
extern "C" void kernel_launch(void* const* d_in, const int* in_sizes, int n_in,
                              void* d_out, int out_size, void* d_ws, size_t ws_size,
                              hipStream_t stream)
{
    const float* A = (const float*)d_in[0];
    const float* B = (const float*)d_in[1];
    float* out = (float*)d_out;
    float* partials = (float*)d_ws;   // (16*16*48) floats = 48 KB of scratch

    dim3 grid(IMG / TILE, IMG / TILE, NIMG);   // 16 x 16 x 48 = 12288 WGs
    nsml_loss_tile<<<grid, 256, 0, stream>>>(A, B, partials);

    const int nPart = (IMG / TILE) * (IMG / TILE) * NIMG;      // 12288
    const float scale = 1.0f / 50331648.0f;                    // 1 / (48*1024*1024)
    nsml_reduce_wmma<<<1, 32, 0, stream>>>(partials, nPart, out, scale);
}